// HeadAttentionMulti_5085241278574
// MI455X (gfx1250) — compile-verified
//
#include <hip/hip_runtime.h>
#include <math.h>

#define B_  8
#define S_  1024
#define H_  16
#define DH_ 64
#define D_  1024

typedef float v2f __attribute__((ext_vector_type(2)));
typedef float v8f __attribute__((ext_vector_type(8)));

// D = A(16x4 f32) * B(4x16 f32) + C(16x16 f32)  -- fp32 matrix pipe
__device__ __forceinline__ v8f wmma_f32(v2f a, v2f b, v8f c) {
  return __builtin_amdgcn_wmma_f32_16x16x4_f32(false, a, false, b, (short)0, c, false, false);
}

// Async global->LDS copy, 16B per lane (GVS mode: SGPR base + per-lane byte offset).
// Tracked by ASYNCcnt; LDS dest address = low 32 bits of the generic pointer.
__device__ __forceinline__ void async_f4(const float* base, int byte_off, void* lds_dst) {
  asm volatile("global_load_async_to_lds_b128 %0, %1, %2"
               :: "v"((unsigned)(uintptr_t)lds_dst),
                  "v"(byte_off),
                  "s"((unsigned long long)(uintptr_t)base)
               : "memory");
}
__device__ __forceinline__ void wait_async() {
  asm volatile("s_wait_asynccnt 0x0" ::: "memory");
}

// ---------------------------------------------------------------------------
// 64x64 block-tile GEMM body: out[64x64] = X[64xK] * W[Kx64] + bias
// 256 threads = 8 waves in a 4(M) x 2(N) grid; each wave: 16x32 via 2 accums.
// ---------------------------------------------------------------------------
template <bool HEAD_SPLIT>
__device__ __forceinline__ void gemm_tile_64x64(
    const float* __restrict__ X, const float* __restrict__ W,
    const float* __restrict__ bias, float* __restrict__ out,
    int mbase, int nbase)
{
  __shared__ float As[64][36];   // pad: conflict-free frag reads, 16B-aligned rows
  __shared__ float Bs[32][68];

  const int t     = threadIdx.x;
  const int wid   = t >> 5;
  const int lane  = t & 31;
  const int wrow  = wid & 3;
  const int wcol  = wid >> 2;
  const int lhalf = lane >> 4;
  const int l15   = lane & 15;

  const int ra = t >> 3, ca = (t & 7) << 2;    // A-stage coords (64x32)
  const int rb = t >> 4, cb = (t & 15) << 2;   // B-stage coords (32x64)

  v8f acc0 = {}; v8f acc1 = {};

  for (int kb = 0; kb < D_; kb += 32) {
    // stage A: 64 rows x 32 cols, async straight into LDS
    async_f4(X, ((mbase + ra)      * D_ + kb + ca) * 4, &As[ra][ca]);
    async_f4(X, ((mbase + ra + 32) * D_ + kb + ca) * 4, &As[ra + 32][ca]);
    // stage B: 32 rows x 64 cols
    async_f4(W, ((kb + rb)      * D_ + nbase + cb) * 4, &Bs[rb][cb]);
    async_f4(W, ((kb + rb + 16) * D_ + nbase + cb) * 4, &Bs[rb + 16][cb]);
    wait_async();
    __syncthreads();

    const int am = wrow * 16 + l15;
    const int n0 = wcol * 32 + l15;
#pragma unroll
    for (int kk = 0; kk < 32; kk += 4) {
      const int ka = kk + (lhalf << 1);          // lanes 16-31 hold K+2,K+3
      v2f a;  a.x  = As[am][ka];       a.y  = As[am][ka + 1];
      v2f b0; b0.x = Bs[ka][n0];       b0.y = Bs[ka + 1][n0];
      acc0 = wmma_f32(a, b0, acc0);
      v2f b1; b1.x = Bs[ka][n0 + 16];  b1.y = Bs[ka + 1][n0 + 16];
      acc1 = wmma_f32(a, b1, acc1);
    }
    __syncthreads();
  }

  // C/D layout: VGPR g -> row g (lanes 0-15) / row g+8 (lanes 16-31), col = lane&15
  const int mrow0 = mbase + wrow * 16 + (lhalf << 3);
#pragma unroll
  for (int j = 0; j < 2; ++j) {
    const v8f acc  = j ? acc1 : acc0;
    const int ncol = nbase + wcol * 32 + j * 16 + l15;
    const float bv = bias[ncol];
#pragma unroll
    for (int g = 0; g < 8; ++g) {
      const int m   = mrow0 + g;
      const float val = acc[g] + bv;
      if (HEAD_SPLIT) {      // (B*S, D) -> (B,H,S,DH)
        const int bi = m >> 10, s = m & 1023;
        const int h  = ncol >> 6, dh = ncol & 63;
        out[((size_t)bi << 20) + ((size_t)h << 16) + (s << 6) + dh] = val;
      } else {               // final projection output: never re-read
        __builtin_nontemporal_store(val, out + (size_t)m * D_ + ncol);
      }
    }
  }
}

__global__ __launch_bounds__(256) void qkv_gemm_kernel(
    const float* __restrict__ x,
    const float* __restrict__ wq, const float* __restrict__ bq,
    const float* __restrict__ wk, const float* __restrict__ bk,
    const float* __restrict__ wv, const float* __restrict__ bv,
    float* __restrict__ qo, float* __restrict__ ko, float* __restrict__ vo)
{
  const float* W; const float* bias; float* out;
  if      (blockIdx.z == 0) { W = wq; bias = bq; out = qo; }
  else if (blockIdx.z == 1) { W = wk; bias = bk; out = ko; }
  else                      { W = wv; bias = bv; out = vo; }
  gemm_tile_64x64<true>(x, W, bias, out, blockIdx.x * 64, blockIdx.y * 64);
}

__global__ __launch_bounds__(256) void out_gemm_kernel(
    const float* __restrict__ z, const float* __restrict__ wo,
    const float* __restrict__ bo, float* __restrict__ out)
{
  gemm_tile_64x64<false>(z, wo, bo, out, blockIdx.x * 64, blockIdx.y * 64);
}

// ---------------------------------------------------------------------------
// Fused logits + mask + softmax per (b, h, 64 q-rows).
// Phase 1: WMMA Q*K^T over 16 k-tiles -> NT-store qk_values, store masked logits.
// Phase 2: per-wave row softmax, 32 values/lane kept in registers (1 rd + 1 wr).
// ---------------------------------------------------------------------------
__global__ __launch_bounds__(256) void attn_softmax_kernel(
    const float* __restrict__ qin, const float* __restrict__ kin,
    const float* __restrict__ mask,
    float* __restrict__ qk_out, float* __restrict__ attn_out)
{
  __shared__ float Qs[64][68];
  __shared__ float Ks[64][68];

  const int b = blockIdx.z, h = blockIdx.y;
  const int qbase = blockIdx.x * 64;
  const int t     = threadIdx.x;
  const int wid   = t >> 5, lane = t & 31;
  const int wrow  = wid & 3, wcol = wid >> 2;
  const int lhalf = lane >> 4, l15 = lane & 15;

  const size_t bh      = (size_t)b * H_ + h;
  const float* qptr    = qin + (bh << 16);   // S*DH = 65536
  const float* kptr    = kin + (bh << 16);
  const size_t logBase = bh << 20;           // S*S

  const int rs = t >> 2, cs = (t & 3) << 4;  // stage coords: 64 rows x 64 cols
  { // stage Q tile (resident for whole kernel)
#pragma unroll
    for (int p = 0; p < 4; ++p)
      async_f4(qptr, ((qbase + rs) * 64 + cs + 4 * p) * 4, &Qs[rs][cs + 4 * p]);
  }

  float mq[8];
  {
    const int qrow0 = qbase + wrow * 16 + (lhalf << 3);
#pragma unroll
    for (int g = 0; g < 8; ++g) mq[g] = mask[b * S_ + qrow0 + g];
  }

  for (int nb = 0; nb < S_; nb += 64) {
    { // stage K tile (64 rows x 64 dh)
#pragma unroll
      for (int p = 0; p < 4; ++p)
        async_f4(kptr, ((nb + rs) * 64 + cs + 4 * p) * 4, &Ks[rs][cs + 4 * p]);
    }
    wait_async();
    __syncthreads();

    v8f acc0 = {}; v8f acc1 = {};
    const int am = wrow * 16 + l15;
    const int bn = wcol * 32 + l15;
#pragma unroll
    for (int kk = 0; kk < 64; kk += 4) {
      const int ka = kk + (lhalf << 1);
      v2f a;  a.x  = Qs[am][ka];           a.y  = Qs[am][ka + 1];
      v2f b0; b0.x = Ks[bn][ka];           b0.y = Ks[bn][ka + 1];        // B = K^T
      acc0 = wmma_f32(a, b0, acc0);
      v2f b1; b1.x = Ks[bn + 16][ka];      b1.y = Ks[bn + 16][ka + 1];
      acc1 = wmma_f32(a, b1, acc1);
    }

    const int qrow0 = qbase + wrow * 16 + (lhalf << 3);
#pragma unroll
    for (int j = 0; j < 2; ++j) {
      const v8f acc  = j ? acc1 : acc0;
      const int ncol = nb + wcol * 32 + j * 16 + l15;
      const float mk = mask[b * S_ + ncol];
#pragma unroll
      for (int g = 0; g < 8; ++g) {
        const int qr      = qrow0 + g;
        const float logit = acc[g] * 0.125f;               // 1/sqrt(DH)
        const size_t idx  = logBase + ((size_t)qr << 10) + ncol;
        __builtin_nontemporal_store(logit, qk_out + idx);  // never re-read
        const float mm = fminf(1.0f, mq[g] + mk);
        attn_out[idx] = fmaf(mm, -1.0e9f, logit);          // masked logits (temp)
      }
    }
    __syncthreads();
  }

  __threadfence();
  __syncthreads();

  // Phase 2: wave 'wid' owns rows [qbase + wid*8, +8); stripe is WGP$/L2-hot.
  for (int i = 0; i < 8; ++i) {
    const int qr = qbase + wid * 8 + i;
    float* rowp  = attn_out + logBase + ((size_t)qr << 10);
    float v[32];
    float mx = -3.0e38f;
#pragma unroll
    for (int jj = 0; jj < 32; ++jj) { v[jj] = rowp[lane + (jj << 5)]; mx = fmaxf(mx, v[jj]); }
#pragma unroll
    for (int off = 16; off > 0; off >>= 1) mx = fmaxf(mx, __shfl_xor(mx, off, 32));
    float sum = 0.0f;
#pragma unroll
    for (int jj = 0; jj < 32; ++jj) { v[jj] = __expf(v[jj] - mx); sum += v[jj]; }
#pragma unroll
    for (int off = 16; off > 0; off >>= 1) sum += __shfl_xor(sum, off, 32);
    const float inv = 1.0f / sum;
#pragma unroll
    for (int jj = 0; jj < 32; ++jj) rowp[lane + (jj << 5)] = v[jj] * inv;
  }
}

// ---------------------------------------------------------------------------
// z = attn @ V per (b,h): M=1024, N=64, K=1024. Stores z as (B,S,D) for proj.
// ---------------------------------------------------------------------------
__global__ __launch_bounds__(256) void av_gemm_kernel(
    const float* __restrict__ attn, const float* __restrict__ vin,
    float* __restrict__ z)
{
  __shared__ float As[64][36];
  __shared__ float Bs[32][68];

  const int b = blockIdx.z, h = blockIdx.y;
  const int mbase = blockIdx.x * 64;
  const int t     = threadIdx.x;
  const int wid   = t >> 5, lane = t & 31;
  const int wrow  = wid & 3, wcol = wid >> 2;
  const int lhalf = lane >> 4, l15 = lane & 15;

  const size_t bh = (size_t)b * H_ + h;
  const float* ap = attn + (bh << 20);
  const float* vp = vin  + (bh << 16);

  const int ra = t >> 3, ca = (t & 7) << 2;
  const int rb = t >> 4, cb = (t & 15) << 2;

  v8f acc0 = {}; v8f acc1 = {};

  for (int kb = 0; kb < S_; kb += 32) {
    async_f4(ap, ((mbase + ra)      * S_ + kb + ca) * 4, &As[ra][ca]);
    async_f4(ap, ((mbase + ra + 32) * S_ + kb + ca) * 4, &As[ra + 32][ca]);
    async_f4(vp, ((kb + rb)      * 64 + cb) * 4, &Bs[rb][cb]);
    async_f4(vp, ((kb + rb + 16) * 64 + cb) * 4, &Bs[rb + 16][cb]);
    wait_async();
    __syncthreads();

    const int am = wrow * 16 + l15;
    const int n0 = wcol * 32 + l15;
#pragma unroll
    for (int kk = 0; kk < 32; kk += 4) {
      const int ka = kk + (lhalf << 1);
      v2f a;  a.x  = As[am][ka];       a.y  = As[am][ka + 1];
      v2f b0; b0.x = Bs[ka][n0];       b0.y = Bs[ka + 1][n0];
      acc0 = wmma_f32(a, b0, acc0);
      v2f b1; b1.x = Bs[ka][n0 + 16];  b1.y = Bs[ka + 1][n0 + 16];
      acc1 = wmma_f32(a, b1, acc1);
    }
    __syncthreads();
  }

  const int mrow0 = mbase + wrow * 16 + (lhalf << 3);
#pragma unroll
  for (int j = 0; j < 2; ++j) {
    const v8f acc = j ? acc1 : acc0;
    const int dh  = wcol * 32 + j * 16 + l15;
#pragma unroll
    for (int g = 0; g < 8; ++g) {
      const int s = mrow0 + g;
      // z[b][s][h*64+dh]
      z[((size_t)b << 20) + ((size_t)s << 10) + (h << 6) + dh] = acc[g];
    }
  }
}

extern "C" void kernel_launch(void* const* d_in, const int* in_sizes, int n_in,
                              void* d_out, int out_size, void* d_ws, size_t ws_size,
                              hipStream_t stream)
{
  (void)in_sizes; (void)n_in; (void)out_size; (void)ws_size;

  const float* x    = (const float*)d_in[0];
  const float* mask = (const float*)d_in[1];
  const float* wq   = (const float*)d_in[2];
  const float* bq   = (const float*)d_in[3];
  const float* wk   = (const float*)d_in[4];
  const float* bk   = (const float*)d_in[5];
  const float* wv   = (const float*)d_in[6];
  const float* bv   = (const float*)d_in[7];
  const float* wo   = (const float*)d_in[8];
  const float* bo   = (const float*)d_in[9];

  float* out = (float*)d_out;
  const size_t OUT_N  = (size_t)B_ * S_ * D_;        // 8,388,608
  const size_t ATTN_N = (size_t)B_ * H_ * S_ * S_;   // 134,217,728
  const size_t QKV_N  = (size_t)B_ * H_ * S_ * DH_;  // 8,388,608
  float* attn = out  + OUT_N;
  float* qk   = attn + ATTN_N;
  float* qo   = qk   + ATTN_N;
  float* ko   = qo   + QKV_N;
  float* vo   = ko   + QKV_N;
  float* z    = (float*)d_ws;                        // 32 MB scratch (B,S,D)

  dim3 blk(256, 1, 1);
  qkv_gemm_kernel    <<<dim3(128, 16, 3), blk, 0, stream>>>(x, wq, bq, wk, bk, wv, bv, qo, ko, vo);
  attn_softmax_kernel<<<dim3(16, 16, 8),  blk, 0, stream>>>(qo, ko, mask, qk, attn);
  av_gemm_kernel     <<<dim3(16, 16, 8),  blk, 0, stream>>>(attn, vo, z);
  out_gemm_kernel    <<<dim3(128, 16, 1), blk, 0, stream>>>(z, wo, bo, out);
}